// Linear_62491774157440
// MI455X (gfx1250) — compile-verified
//
#include <hip/hip_runtime.h>

#define GLN_SIZE   1024
#define GLN_INPUT  1024
#define GLN_CTXSZ  512
#define GLN_CMS    6
#define GLN_NCTX   64
#define GLN_BATCH  128
#define GLN_LR     0.01f
#define GLN_OCLIP  0.01f
#define GLN_WCLIP  5.0f
#define PROJ_ROWS  (GLN_SIZE * GLN_CMS)   // 6144

typedef float v2f __attribute__((ext_vector_type(2)));
typedef float v4f __attribute__((ext_vector_type(4)));
typedef float v8f __attribute__((ext_vector_type(8)));

// ---------------------------------------------------------------------------
// Kernel 0: transpose logits (1024x128) -> logitsT (128x1024), LDS-tiled.
// Makes every later logits access coalesced instead of 512B-strided.
// ---------------------------------------------------------------------------
__global__ __launch_bounds__(256) void transpose_logits(const float* __restrict__ in,
                                                        float* __restrict__ out) {
  __shared__ float tilebuf[32][33];
  const int tx = threadIdx.x & 31, ty = threadIdx.x >> 5;  // ty in [0,8)
  const int ti = blockIdx.x / (GLN_BATCH / 32);            // 32 tiles over INPUT
  const int tj = blockIdx.x % (GLN_BATCH / 32);            // 4 tiles over BATCH
#pragma unroll
  for (int r = 0; r < 4; ++r)
    tilebuf[ty + 8 * r][tx] =
        in[(size_t)(ti * 32 + ty + 8 * r) * GLN_BATCH + tj * 32 + tx];
  __syncthreads();
#pragma unroll
  for (int r = 0; r < 4; ++r)
    out[(size_t)(tj * 32 + ty + 8 * r) * GLN_INPUT + ti * 32 + tx] =
        tilebuf[tx][ty + 8 * r];
}

// ---------------------------------------------------------------------------
// Kernel 1: projected = P(6144x512) @ X(512x128), fp32 via V_WMMA_F32_16X16X4_F32
// One wave per 16x16 output tile, 4 waves/block. 3072 tiles -> 768 blocks.
// ---------------------------------------------------------------------------
__global__ __launch_bounds__(128) void proj_gemm(const float* __restrict__ P,
                                                 const float* __restrict__ X,
                                                 float* __restrict__ proj) {
  const int wave = threadIdx.x >> 5;
  const int lane = threadIdx.x & 31;
  const int tile = blockIdx.x * 4 + wave;
  const int tiles_n = GLN_BATCH / 16;            // 8
  const int m0 = (tile / tiles_n) * 16;
  const int n0 = (tile % tiles_n) * 16;
  const int half = lane >> 4;                    // 0: K pair {0,1}, 1: K pair {2,3}
  const int l16  = lane & 15;

  v8f acc = {};
  const float* __restrict__ arow = P + (size_t)(m0 + l16) * GLN_CTXSZ + 2 * half;
  const float* __restrict__ bcol = X + n0 + l16;

  for (int k0 = 0; k0 < GLN_CTXSZ; k0 += 4) {
    v2f a, b;
    a.x = arow[k0];                              // K = k0   + 2*half
    a.y = arow[k0 + 1];                          // K = k0+1 + 2*half
    b.x = bcol[(size_t)(k0 + 2 * half) * GLN_BATCH];
    b.y = bcol[(size_t)(k0 + 1 + 2 * half) * GLN_BATCH];
    acc = __builtin_amdgcn_wmma_f32_16x16x4_f32(false, a, false, b,
                                                (short)0, acc, false, false);
  }
#pragma unroll
  for (int r = 0; r < 8; ++r)
    proj[(size_t)(m0 + r + 8 * half) * GLN_BATCH + n0 + l16] = acc[r];
}

// ---------------------------------------------------------------------------
// Kernel 2: ctx_idx[s,b] = sum_m ((proj[s,m,b] > pbias[s,m]) << m)
// ---------------------------------------------------------------------------
__global__ void ctx_hash(const float* __restrict__ proj,
                         const float* __restrict__ pbias,
                         int* __restrict__ ctx) {
  int t = blockIdx.x * blockDim.x + threadIdx.x;
  if (t >= GLN_SIZE * GLN_BATCH) return;
  int s = t >> 7, b = t & 127;
  int v = 0;
#pragma unroll
  for (int m = 0; m < GLN_CMS; ++m) {
    float p  = proj[(size_t)(s * GLN_CMS + m) * GLN_BATCH + b];
    float pb = pbias[s * GLN_CMS + m];
    v |= (p > pb) ? (1 << m) : 0;
  }
  ctx[t] = v;
}

// ---------------------------------------------------------------------------
// Kernel 3: per-neuron block (grid=1024, block=256 = 8 waves).
//   - gathered dots (coalesced v4f from logitsT) -> out, sigmoid -> g[b]
//   - per-context winner (atomicMax = last batch wins, matches scatter .set)
//   - stream 64 weight rows: copy or update+clip, nontemporal (256MB > L2)
// ---------------------------------------------------------------------------
__global__ __launch_bounds__(256) void gln_main(const float* __restrict__ lt,
                                                const float* __restrict__ targets,
                                                const float* __restrict__ weights,
                                                const int* __restrict__ ctx_idx,
                                                const float* __restrict__ bias_p,
                                                float* __restrict__ out,
                                                float* __restrict__ new_w) {
  __shared__ int   ctx[GLN_BATCH];
  __shared__ int   winner[GLN_NCTX];
  __shared__ float g[GLN_BATCH];

  const int s = blockIdx.x;
  const int t = threadIdx.x;
  const int wave = t >> 5, lane = t & 31;

  if (t < GLN_NCTX) winner[t] = -1;
  __syncthreads();
  if (t < GLN_BATCH) {
    int c = ctx_idx[s * GLN_BATCH + t];
    ctx[t] = c;
    atomicMax(&winner[c], t);                    // last (max) batch index wins
  }
  __syncthreads();

  const float bias = *bias_p;

  // gathered dot products: wave w handles batches w*16 .. w*16+15
  for (int j = 0; j < 16; ++j) {
    int b = wave * 16 + j;
    int c = ctx[b];
    const v4f* __restrict__ wr4 =
        (const v4f*)(weights + ((size_t)s * GLN_NCTX + c) * GLN_INPUT);
    const v4f* __restrict__ lt4 = (const v4f*)(lt + (size_t)b * GLN_INPUT);
    float sum = 0.f;
#pragma unroll
    for (int kk = lane; kk < GLN_INPUT / 4; kk += 32) {   // 8 iters, coalesced
      v4f w = wr4[kk], x = lt4[kk];
      sum += w.x * x.x + w.y * x.y + w.z * x.z + w.w * x.w;
    }
#pragma unroll
    for (int off = 16; off; off >>= 1) sum += __shfl_xor(sum, off, 32);
    if (lane == 0) {
      float val = (s == 0) ? bias : sum;         // bias neuron overwrites row 0
      out[s * GLN_BATCH + b] = val;
      float sg = 1.f / (1.f + expf(-val));
      sg = fminf(fmaxf(sg, GLN_OCLIP), 1.f - GLN_OCLIP);
      g[b] = GLN_LR * (sg - targets[b]);
    }
  }
  __syncthreads();

  // weight copy / update: 64 rows of 1024 floats; one v4f per thread per row
  const size_t base = (size_t)s * GLN_NCTX * GLN_INPUT;
  for (int c = 0; c < GLN_NCTX; ++c) {
    const v4f* __restrict__ src = (const v4f*)(weights + base + (size_t)c * GLN_INPUT);
    v4f* __restrict__ dst = (v4f*)(new_w + base + (size_t)c * GLN_INPUT);
    v4f w4 = __builtin_nontemporal_load(&src[t]);
    int bw = winner[c];
    if (bw >= 0) {
      float gb = g[bw];
      v4f l4 = *((const v4f*)(lt + (size_t)bw * GLN_INPUT) + t);  // coalesced
      w4.x = fminf(fmaxf(w4.x - gb * l4.x, -GLN_WCLIP), GLN_WCLIP);
      w4.y = fminf(fmaxf(w4.y - gb * l4.y, -GLN_WCLIP), GLN_WCLIP);
      w4.z = fminf(fmaxf(w4.z - gb * l4.z, -GLN_WCLIP), GLN_WCLIP);
      w4.w = fminf(fmaxf(w4.w - gb * l4.w, -GLN_WCLIP), GLN_WCLIP);
    }
    __builtin_nontemporal_store(w4, &dst[t]);
  }
}

// ---------------------------------------------------------------------------
extern "C" void kernel_launch(void* const* d_in, const int* in_sizes, int n_in,
                              void* d_out, int out_size, void* d_ws, size_t ws_size,
                              hipStream_t stream) {
  const float* logits   = (const float*)d_in[0];  // (1024,128)
  const float* ctx_in   = (const float*)d_in[1];  // (512,128)
  const float* targets  = (const float*)d_in[2];  // (128,)
  const float* proj_mat = (const float*)d_in[3];  // (1024,6,512)
  const float* proj_b   = (const float*)d_in[4];  // (1024,6,1)
  const float* weights  = (const float*)d_in[5];  // (1024,64,1024)
  const float* bias     = (const float*)d_in[6];  // scalar

  float* out   = (float*)d_out;                        // (1024,128)
  float* new_w = out + (size_t)GLN_SIZE * GLN_BATCH;   // (1024,64,1024)

  float* proj = (float*)d_ws;                                    // 6144*128 f
  int*   cidx = (int*)(proj + (size_t)PROJ_ROWS * GLN_BATCH);    // 1024*128 i
  float* lt   = (float*)(cidx + (size_t)GLN_SIZE * GLN_BATCH);   // 128*1024 f

  // 0) transpose logits for coalesced access in the hot kernel
  transpose_logits<<<(GLN_INPUT / 32) * (GLN_BATCH / 32), 256, 0, stream>>>(
      logits, lt);

  // 1) fp32 WMMA projection GEMM: 3072 tiles, 4 waves/block
  proj_gemm<<<(PROJ_ROWS / 16) * (GLN_BATCH / 16) / 4, 128, 0, stream>>>(
      proj_mat, ctx_in, proj);

  // 2) context hashing
  ctx_hash<<<(GLN_SIZE * GLN_BATCH + 255) / 256, 256, 0, stream>>>(
      proj, proj_b, cidx);

  // 3) fused output + update + full weight scatter-copy (bandwidth-bound)
  gln_main<<<GLN_SIZE, 256, 0, stream>>>(lt, targets, weights, cidx,
                                         bias, out, new_w);
}